// SSIMLoss_30399778521183
// MI455X (gfx1250) — compile-verified
//
#include <hip/hip_runtime.h>

typedef __attribute__((ext_vector_type(2))) float v2f;
typedef __attribute__((ext_vector_type(8))) float v8f;

#define IMG      512
#define TILE     32            // 32x32 output tile per workgroup
#define IN       42            // TILE + 11 - 1 (haloed input span)
#define RAWSTR   44            // padded LDS stride for raw tiles
#define HPROWS   44            // 42 live rows + 2 zero rows (so WMMA K=28 never clamps)
#define HPSTR    48            // padded stride: rows 2 apart land 32 banks apart -> conflict free
#define NTHR     128           // 4 wave32
#define NPIX     12582912.0f   // 16*3*512*512

// Normalized 1-D Gaussian, sigma=1.5, 11 taps (compile-time literals for pass 1)
#define G0 0.001028380f
#define G1 0.007598662f
#define G2 0.036000773f
#define G3 0.109360667f
#define G4 0.213005529f
#define G5 0.266011724f

__global__ __launch_bounds__(NTHR)
void ssim_tile_kernel(const float* __restrict__ sr, const float* __restrict__ hr,
                      float* __restrict__ partials) {
    __shared__ float lds_raw[2][IN][RAWSTR];      // sr, hr haloed tiles
    __shared__ float lds_hp[5][HPROWS][HPSTR];    // horizontal conv of 5 signals
    __shared__ float lds_gtab[48];                // gtab[d+15] = g[d] for d in [0,10], else 0
    __shared__ float lds_red[NTHR];

    const int tid  = threadIdx.x;
    const int tr   = blockIdx.x >> 4;             // 16x16 tiles per 512x512 plane
    const int tc   = blockIdx.x & 15;
    const int r0   = tr * TILE;
    const int c0   = tc * TILE;
    const size_t poff = (size_t)blockIdx.y * (size_t)(IMG * IMG);

    // Gaussian lookup table for building the Toeplitz A operand per-lane.
    if (tid < 48) {
        float d = (float)(tid - 20);
        float v = expf(-d * d * (1.0f / 4.5f)) * G5;   // exp(-d^2/(2*1.5^2)) / sum
        lds_gtab[tid] = (tid >= 15 && tid <= 25) ? v : 0.0f;
    }

    // ---- Stage haloed sr/hr tiles (zero padding outside the image) ----
    for (int idx = tid; idx < IN * IN; idx += NTHR) {
        int i  = idx / IN;
        int jj = idx - i * IN;
        int gr = r0 - 5 + i;
        int gc = c0 - 5 + jj;
        bool ok = (gr >= 0) & (gr < IMG) & (gc >= 0) & (gc < IMG);
        size_t off = poff + (size_t)gr * IMG + gc;
        lds_raw[0][i][jj] = ok ? sr[off] : 0.0f;
        lds_raw[1][i][jj] = ok ? hr[off] : 0.0f;
    }
    __syncthreads();

    // ---- Pass 1: horizontal 11-tap conv of the 5 signals (VALU) ----
    for (int idx = tid; idx < HPROWS * TILE; idx += NTHR) {
        int i = idx >> 5;          // row in [0,44)
        int j = idx & 31;          // output col in [0,32)
        float m1 = 0.f, m2 = 0.f, s11 = 0.f, s22 = 0.f, s12 = 0.f;
        if (i < IN) {
            const float gw[11] = { G0,G1,G2,G3,G4,G5,G4,G3,G2,G1,G0 };
#pragma unroll
            for (int t = 0; t < 11; ++t) {
                float sv = lds_raw[0][i][j + t];
                float hv = lds_raw[1][i][j + t];
                float w  = gw[t];                 // compile-time literal after unroll
                float ws = w * sv;
                float wh = w * hv;
                m1  += ws;
                m2  += wh;
                s11 += ws * sv;
                s22 += wh * hv;
                s12 += ws * hv;
            }
        }
        lds_hp[0][i][j] = m1;
        lds_hp[1][i][j] = m2;
        lds_hp[2][i][j] = s11;
        lds_hp[3][i][j] = s22;
        lds_hp[4][i][j] = s12;
    }
    __syncthreads();

    // ---- Pass 2: vertical conv as Toeplitz matmul on the matrix pipe ----
    // Out(16x16) = A(16x28) * B(28x16), A[m][k] = g[k-m], B[k][n] = Hp[R0+k][C0+n]
    const int lane = tid & 31;
    const int wave = tid >> 5;
    const int half = lane >> 4;       // K-half selector for A/B operands
    const int n    = lane & 15;       // column (B/C/D N index)
    const int R0   = (wave >> 1) * 16;
    const int C0   = (wave & 1) * 16;

    // A operand slices (shared by all 5 signals): lane holds rows m=lane&15,
    // VGPR0 -> K = 4i + 2*half, VGPR1 -> K+1  (ISA 16x4 fp32 A layout)
    v2f aw[7];
#pragma unroll
    for (int i = 0; i < 7; ++i) {
        int k = 4 * i + 2 * half;
        aw[i].x = lds_gtab[k     - n + 15];   // n == lane&15 == A row m here
        aw[i].y = lds_gtab[k + 1 - n + 15];
    }

#define VCONV(ACC, SIDX)                                                        \
    do {                                                                        \
        _Pragma("unroll")                                                       \
        for (int i7 = 0; i7 < 7; ++i7) {                                        \
            int rb = R0 + 4 * i7 + 2 * half;                                    \
            v2f b;                                                              \
            b.x = lds_hp[SIDX][rb][C0 + n];                                     \
            b.y = lds_hp[SIDX][rb + 1][C0 + n];                                 \
            ACC = __builtin_amdgcn_wmma_f32_16x16x4_f32(                        \
                false, aw[i7], false, b, (short)0, ACC, false, false);          \
        }                                                                       \
    } while (0)

    v8f acc_mu1 = {}, acc_mu2 = {}, acc_x11 = {}, acc_x22 = {}, acc_x12 = {};
    VCONV(acc_mu1, 0);
    VCONV(acc_mu2, 1);
    VCONV(acc_x11, 2);
    VCONV(acc_x22, 3);
    VCONV(acc_x12, 4);
#undef VCONV

    // ---- Per-element SSIM map + local sum (8 elements per lane) ----
    const float c1 = 0.0001f;   // (0.01*1)^2
    const float c2 = 0.0009f;   // (0.03*1)^2
    float lsum = 0.0f;
#pragma unroll
    for (int v = 0; v < 8; ++v) {
        float mu1 = acc_mu1[v];
        float mu2 = acc_mu2[v];
        float mu1sq = mu1 * mu1;
        float mu2sq = mu2 * mu2;
        float mu12  = mu1 * mu2;
        float sg1 = acc_x11[v] - mu1sq;
        float sg2 = acc_x22[v] - mu2sq;
        float sg12 = acc_x12[v] - mu12;
        float num = (2.0f * mu12 + c1) * (2.0f * sg12 + c2);
        float den = (mu1sq + mu2sq + c1) * (sg1 + sg2 + c2) + 1e-12f;
        lsum += num / den;
    }

    // ---- Deterministic block reduction ----
    lds_red[tid] = lsum;
    __syncthreads();
#pragma unroll
    for (int o = NTHR / 2; o > 0; o >>= 1) {
        if (tid < o) lds_red[tid] += lds_red[tid + o];
        __syncthreads();
    }
    if (tid == 0)
        partials[(size_t)blockIdx.y * gridDim.x + blockIdx.x] = lds_red[0];
}

__global__ __launch_bounds__(256)
void ssim_reduce_kernel(const float* __restrict__ partials, int n, float* __restrict__ out) {
    __shared__ float red[256];
    float s = 0.0f;
    for (int i = threadIdx.x; i < n; i += 256) s += partials[i];
    red[threadIdx.x] = s;
    __syncthreads();
#pragma unroll
    for (int o = 128; o > 0; o >>= 1) {
        if (threadIdx.x < o) red[threadIdx.x] += red[threadIdx.x + o];
        __syncthreads();
    }
    if (threadIdx.x == 0) out[0] = 1.0f - red[0] * (1.0f / NPIX);
}

extern "C" void kernel_launch(void* const* d_in, const int* in_sizes, int n_in,
                              void* d_out, int out_size, void* d_ws, size_t ws_size,
                              hipStream_t stream) {
    (void)in_sizes; (void)n_in; (void)out_size; (void)ws_size;
    const float* sr = (const float*)d_in[0];
    const float* hr = (const float*)d_in[1];
    float* out      = (float*)d_out;
    float* partials = (float*)d_ws;              // 256*48 floats = 48 KB

    dim3 grid(256, 48);                          // 16x16 tiles per plane, 48 planes
    ssim_tile_kernel<<<grid, NTHR, 0, stream>>>(sr, hr, partials);
    ssim_reduce_kernel<<<1, 256, 0, stream>>>(partials, 256 * 48, out);
}